// DDE_62491774157489
// MI455X (gfx1250) — compile-verified
//
#include <hip/hip_runtime.h>

typedef float v2f __attribute__((ext_vector_type(2)));
typedef float v8f __attribute__((ext_vector_type(8)));

#define WPB 8  // waves per block (wave32)

// ---------------- CSR construction ----------------

__global__ void k_zero(int* a, int n) {
    int i = blockIdx.x * blockDim.x + threadIdx.x;
    if (i < n) a[i] = 0;
}

__global__ void k_count(const int* __restrict__ src, const int* __restrict__ dst,
                        int E_, int* deg_f, int* deg_r) {
    int e = blockIdx.x * blockDim.x + threadIdx.x;
    if (e < E_) {
        atomicAdd(&deg_f[dst[e]], 1);
        atomicAdd(&deg_r[src[e]], 1);
    }
}

// single-block exclusive-prefix into off[0..n]; each thread owns a contiguous chunk
__global__ void k_scan(const int* __restrict__ deg, int* __restrict__ off, int n) {
    __shared__ int tsum[1024];
    const int t = threadIdx.x;
    const int C = (n + 1023) / 1024;
    const int begin = t * C;
    int s = 0;
    for (int j = 0; j < C; ++j) {
        int i = begin + j;
        if (i < n) s += deg[i];
    }
    tsum[t] = s;
    __syncthreads();
    for (int d = 1; d < 1024; d <<= 1) {
        int v = (t >= d) ? tsum[t - d] : 0;
        __syncthreads();
        tsum[t] += v;
        __syncthreads();
    }
    int run = tsum[t] - s;  // exclusive prefix of this thread's chunk
    for (int j = 0; j < C; ++j) {
        int i = begin + j;
        if (i < n) { run += deg[i]; off[i + 1] = run; }
    }
    if (t == 0) off[0] = 0;
}

__global__ void k_prep(const int* __restrict__ deg, const int* __restrict__ off,
                       int* cur, float* inv, int n) {
    int i = blockIdx.x * blockDim.x + threadIdx.x;
    if (i < n) {
        int d = deg[i];
        inv[i] = d > 0 ? 1.0f / (float)d : 0.0f;
        cur[i] = off[i];
    }
}

__global__ void k_fill(const int* __restrict__ src, const int* __restrict__ dst, int E_,
                       int* cur_f, int* col_f, int* cur_r, int* col_r) {
    int e = blockIdx.x * blockDim.x + threadIdx.x;
    if (e < E_) {
        int s = src[e], d = dst[e];
        col_f[atomicAdd(&cur_f[d], 1)] = s;
        col_r[atomicAdd(&cur_r[s], 1)] = d;
    }
}

// ---------------- WMMA SpMM gather round ----------------
// One wave per 16-node tile. Scalar K-loop over chunks of 4 CSR edges:
//   D(16x16) += A(16x4 one-hot ownership) x B(4x16 src features), x4 feature tiles.
// Ownership test per lane is just lo <= g < hi against that lane's own row bounds.
__global__ __launch_bounds__(256) void k_gather_wmma(
    const float* __restrict__ hin, const int* __restrict__ off,
    const int* __restrict__ col, const float* __restrict__ inv,
    float* __restrict__ hout, int n)
{
    __shared__ int soff[WPB][17];
    const int lane = threadIdx.x & 31;
    const int w    = threadIdx.x >> 5;
    // wave-uniform scalars (readfirstlane so branches/loop control go to SALU)
    const int nodeBase = __builtin_amdgcn_readfirstlane(
        (blockIdx.x * WPB + w) * 16);

    if (nodeBase < n && lane < 17) {
        int idx = nodeBase + lane;
        soff[w][lane] = off[idx <= n ? idx : n];
    }
    __syncthreads();
    if (nodeBase >= n) return;  // scalar, uniform exit: EXEC all-ones below

    const int half = lane >> 4;   // 0: owns K=0,1   1: owns K=2,3  (A and B layouts)
    const int fl   = lane & 15;   // M for A, feature N for B/C/D

    // per-lane CSR row bounds for node M = fl
    const int lo = soff[w][fl];
    const int hi = soff[w][fl + 1];
    // scalar row range of the whole tile
    const int e0     = __builtin_amdgcn_readfirstlane(soff[w][0]);
    const int eEnd   = __builtin_amdgcn_readfirstlane(soff[w][16]);
    const int nedges = eEnd - e0;
    const int eLast  = eEnd - 1;

    v8f acc0 = {}, acc1 = {}, acc2 = {}, acc3 = {};

#pragma unroll 2
    for (int p = 0; p < nedges; p += 4) {   // scalar loop: s_cmp/s_cbranch
        const int gA = e0 + p + 2 * half;
        const int gB = gA + 1;

        // A[m][k] = 1 iff CSR position g falls inside node m's row
        v2f a;
        a.x = (gA >= lo && gA < hi) ? 1.0f : 0.0f;
        a.y = (gB >= lo && gB < hi) ? 1.0f : 0.0f;

        // clamped safe column loads (finite data, zeroed by A when out of range)
        const int sA = col[gA <= eLast ? gA : eLast];
        const int sB = col[gB <= eLast ? gB : eLast];
        const int oA = sA * 64 + fl;
        const int oB = sB * 64 + fl;

        v2f b0, b1, b2, b3;
        b0.x = hin[oA];      b0.y = hin[oB];
        b1.x = hin[oA + 16]; b1.y = hin[oB + 16];
        b2.x = hin[oA + 32]; b2.y = hin[oB + 32];
        b3.x = hin[oA + 48]; b3.y = hin[oB + 48];

        acc0 = __builtin_amdgcn_wmma_f32_16x16x4_f32(false, a, false, b0, (short)0, acc0, false, false);
        acc1 = __builtin_amdgcn_wmma_f32_16x16x4_f32(false, a, false, b1, (short)0, acc1, false, false);
        acc2 = __builtin_amdgcn_wmma_f32_16x16x4_f32(false, a, false, b2, (short)0, acc2, false, false);
        acc3 = __builtin_amdgcn_wmma_f32_16x16x4_f32(false, a, false, b3, (short)0, acc3, false, false);
    }

    // scale rows by 1/deg and store: C/D layout -> VGPR v holds M = v + 8*half
    if (nodeBase + 16 <= n) {               // scalar fast path (always for N%16==0)
#pragma unroll
        for (int v = 0; v < 8; ++v) {
            const int node = nodeBase + v + half * 8;
            const float s = inv[node];
            float* orow = hout + node * 64 + fl;
            orow[0]  = acc0[v] * s;
            orow[16] = acc1[v] * s;
            orow[32] = acc2[v] * s;
            orow[48] = acc3[v] * s;
        }
    } else {
#pragma unroll
        for (int v = 0; v < 8; ++v) {
            const int node = nodeBase + v + half * 8;
            if (node < n) {
                const float s = inv[node];
                float* orow = hout + node * 64 + fl;
                orow[0]  = acc0[v] * s;
                orow[16] = acc1[v] * s;
                orow[32] = acc2[v] * s;
                orow[48] = acc3[v] * s;
            }
        }
    }
}

// ---------------- driver ----------------

extern "C" void kernel_launch(void* const* d_in, const int* in_sizes, int n_in,
                              void* d_out, int out_size, void* d_ws, size_t ws_size,
                              hipStream_t stream) {
    const int N_ = in_sizes[0] / 64;
    const int E_ = in_sizes[1] / 2;
    const float* x   = (const float*)d_in[0];
    const int*   src = (const int*)d_in[1];
    const int*   dst = src + E_;
    float* out = (float*)d_out;

    int* p = (int*)d_ws;
    int* deg_f = p;                p += N_;
    int* deg_r = p;                p += N_;
    int* off_f = p;                p += N_ + 1;
    int* off_r = p;                p += N_ + 1;
    int* cur_f = p;                p += N_;
    int* cur_r = p;                p += N_;
    int* col_f = p;                p += E_;
    int* col_r = p;                p += E_;
    float* inv_f = (float*)p;      p += N_;
    float* inv_r = (float*)p;

    const int TB = 256;
    k_zero <<<(2 * N_ + TB - 1) / TB, TB, 0, stream>>>(deg_f, 2 * N_);
    k_count<<<(E_ + TB - 1) / TB, TB, 0, stream>>>(src, dst, E_, deg_f, deg_r);
    k_scan <<<1, 1024, 0, stream>>>(deg_f, off_f, N_);
    k_scan <<<1, 1024, 0, stream>>>(deg_r, off_r, N_);
    k_prep <<<(N_ + TB - 1) / TB, TB, 0, stream>>>(deg_f, off_f, cur_f, inv_f, N_);
    k_prep <<<(N_ + TB - 1) / TB, TB, 0, stream>>>(deg_r, off_r, cur_r, inv_r, N_);
    k_fill <<<(E_ + TB - 1) / TB, TB, 0, stream>>>(src, dst, E_, cur_f, col_f, cur_r, col_r);

    const int tiles  = (N_ + 15) / 16;
    const int blocks = (tiles + WPB - 1) / WPB;

    const float* h = x;
    for (int r = 0; r < 4; ++r) {
        float* o = out + (size_t)r * (size_t)N_ * 64;
        k_gather_wmma<<<blocks, 32 * WPB, 0, stream>>>(h, off_f, col_f, inv_f, o, N_);
        h = o;
    }
    h = x;
    for (int r = 0; r < 4; ++r) {
        float* o = out + (size_t)(4 + r) * (size_t)N_ * 64;
        k_gather_wmma<<<blocks, 32 * WPB, 0, stream>>>(h, off_r, col_r, inv_r, o, N_);
        h = o;
    }
}